// GLCMTextureDetection_13331578487480
// MI455X (gfx1250) — compile-verified
//
#include <hip/hip_runtime.h>
#include <hip/hip_bf16.h>
#include <stdint.h>

#define LEVELS 256
#define DIST   5
#define H      512
#define W      512
#define NPIX   (H * W)
#define NPAIRS (H * (W - DIST))
#define NBINS  (LEVELS * LEVELS)
#define BLOCK  1024

typedef __attribute__((ext_vector_type(2))) float v2f;
typedef __attribute__((ext_vector_type(8))) float v8f;

// LDS partition (dynamic, 278.7KB of the 320KB WGP LDS):
//   [0, 256KB)            : uint32 hist[65536]    per-image GLCM histogram
//   [256KB, +16KB)        : float  qbuf[256*16]   Q = P * L_pad   (WMMA stage A out)
//   [+16KB, +128B)        : float  red[32]        red[0..2] band sums, red[8..16] M3
#define HIST_BYTES (NBINS * 4)
#define QBUF_BYTES (LEVELS * 16 * 4)
#define RED_BYTES  (32 * 4)
#define SMEM_BYTES (HIST_BYTES + QBUF_BYTES + RED_BYTES)

__device__ __forceinline__ int gray_at(const float* __restrict__ imgb, int r, int c) {
    const int HWp = NPIX;
    float q0 = fminf(fmaxf(floorf(imgb[0 * HWp + r * W + c] * 255.0f), 0.0f), 255.0f);
    float q1 = fminf(fmaxf(floorf(imgb[1 * HWp + r * W + c] * 255.0f), 0.0f), 255.0f);
    float q2 = fminf(fmaxf(floorf(imgb[2 * HWp + r * W + c] * 255.0f), 0.0f), 255.0f);
    float g  = 0.299f * q0 + 0.587f * q1 + 0.114f * q2;
    return (int)fminf(fmaxf(rintf(g), 0.0f), 255.0f);
}

// symmetrized, normalized GLCM probability
__device__ __forceinline__ float pval(const uint32_t* hist, int r, int c, float invTot) {
    return (float)(hist[(r << 8) | c] + hist[(c << 8) | r]) * invTot;
}

__global__ void __launch_bounds__(BLOCK)
glcm_texture_kernel(const float* __restrict__ img, float* __restrict__ out) {
    extern __shared__ unsigned char smem_raw[];
    uint32_t* hist = (uint32_t*)smem_raw;
    float*    qbuf = (float*)(smem_raw + HIST_BYTES);
    float*    red  = (float*)(smem_raw + HIST_BYTES + QBUF_BYTES);

    const int tid = threadIdx.x;
    const int b   = blockIdx.x;
    const float* imgb = img + (size_t)b * 3 * NPIX;
    const float invTot = 1.0f / (2.0f * (float)NPAIRS);

    // ---- Phase 0: zero histogram + reduction cells ----
    for (int idx = tid; idx < NBINS; idx += BLOCK) hist[idx] = 0u;
    if (tid < 32) red[tid] = 0.0f;
    __syncthreads();

    // ---- Phase 1: GLCM histogram via LDS atomics (320KB WGP LDS) ----
    for (int p = tid; p < NPIX; p += BLOCK) {
        int r = p >> 9;
        int c = p & (W - 1);
        if (c < W - DIST) {
            int gi = gray_at(imgb, r, c);
            int gj = gray_at(imgb, r, c + DIST);
            atomicAdd(&hist[(gi << 8) | gj], 1u);
        }
    }
    __syncthreads();

    // ---- Phase 2: per-bin band features (|i-j| weights, energy) — VALU pass ----
    {
        float s_d = 0.0f, s_h = 0.0f, s_e = 0.0f;
        for (int idx = tid; idx < NBINS; idx += BLOCK) {
            int i = idx >> 8;
            int j = idx & 255;
            float p = (float)(hist[idx] + hist[(j << 8) | i]) * invTot;
            float di = (float)(i - j);
            s_d += p * fabsf(di);
            s_h += p / (1.0f + di * di);
            s_e += p * p;
        }
        atomicAdd(&red[0], s_d);
        atomicAdd(&red[1], s_h);
        atomicAdd(&red[2], s_e);
    }

    // Branchless Vandermonde basis: L[k][n] = c0 + k*(c1 + c2*k), coeffs one-hot in n.
    // Hoisted out of the k-loops so the WMMA inner loop has no EXEC divergence.
    {
        const int lane = tid & 31;
        const int m    = lane & 15;
        const int koff = (lane >> 4) << 1;   // A/B layout: VGPR0=K0/K2, VGPR1=K1/K3
        const float c0 = (m == 0) ? 1.0f : 0.0f;
        const float c1 = (m == 1) ? 1.0f : 0.0f;
        const float c2 = (m == 2) ? 1.0f : 0.0f;

        // ---- Phase 3 (WMMA stage A): Q = P(256x256) x L_pad(256x16), 16 waves ----
        if (tid < 512) {
            const int wid  = tid >> 5;       // row-tile 0..15
            const int rowm = wid * 16 + m;
            v8f acc = {};
            for (int kt = 0; kt < 64; ++kt) {
                int kb = kt * 4 + koff;
                float k0 = (float)kb, k1 = (float)(kb + 1);
                v2f a, bb;
                a.x  = pval(hist, rowm, kb,     invTot);
                a.y  = pval(hist, rowm, kb + 1, invTot);
                bb.x = c0 + k0 * (c1 + c2 * k0);   // B lane = column n (= lane&15)
                bb.y = c0 + k1 * (c1 + c2 * k1);
                acc = __builtin_amdgcn_wmma_f32_16x16x4_f32(
                    false, a, false, bb, (short)0, acc, false, false);
            }
            // D layout: lanes<16 -> M = r, lanes>=16 -> M = 8+r; N = lane&15
            const int drow = wid * 16 + ((lane >> 4) ? 8 : 0);
            const int dcol = lane & 15;
            for (int r = 0; r < 8; ++r) qbuf[(drow + r) * 16 + dcol] = acc[r];
        }
        __syncthreads();

        // ---- Phase 4 (WMMA stage B): M3 = L_pad^T(16x256) x Q(256x16), wave 0 ----
        if (tid < 32) {
            v8f acc = {};
            for (int kt = 0; kt < 64; ++kt) {
                int kb = kt * 4 + koff;
                float k0 = (float)kb, k1 = (float)(kb + 1);
                v2f a, bb;
                a.x  = c0 + k0 * (c1 + c2 * k0);   // A[m][k] = L[k][m]
                a.y  = c0 + k1 * (c1 + c2 * k1);
                bb.x = qbuf[kb       * 16 + m];
                bb.y = qbuf[(kb + 1) * 16 + m];
                acc = __builtin_amdgcn_wmma_f32_16x16x4_f32(
                    false, a, false, bb, (short)0, acc, false, false);
            }
            if (lane < 3) {                  // M3[0..2][0..2]; N=lane, M=r (lanes<16)
                for (int r = 0; r < 3; ++r) red[8 + r * 3 + lane] = acc[r];
            }
        }
    }
    __syncthreads();

    // ---- Phase 5: assemble features, normalize, emit ----
    if (tid == 0) {
        const float S_i  = red[8 + 1 * 3 + 0];
        const float S_j  = red[8 + 0 * 3 + 1];
        const float S_ii = red[8 + 2 * 3 + 0];
        const float S_jj = red[8 + 0 * 3 + 2];
        const float S_ij = red[8 + 1 * 3 + 1];

        float contrast = S_ii - 2.0f * S_ij + S_jj;
        float dissim   = red[0];
        float homog    = red[1];
        float energy   = sqrtf(red[2]);
        float var_i    = S_ii - S_i * S_i;
        float var_j    = S_jj - S_j * S_j;
        float cov      = S_ij - S_i * S_j;
        float stdv     = sqrtf(var_i * var_j);
        float corr     = (stdv < 1e-15f) ? 1.0f : cov / fmaxf(stdv, 1e-15f);

        float feats[5] = {contrast, dissim, homog, energy, corr};
        float fmin = feats[0], fmax = feats[0];
        for (int f = 1; f < 5; ++f) {
            fmin = fminf(fmin, feats[f]);
            fmax = fmaxf(fmax, feats[f]);
        }
        float* ob = out + (size_t)b * 15;
        for (int f = 0; f < 5; ++f) {
            float fn = (feats[f] - fmin) / (fmax - fmin);
            float v  = floorf(fn * 255.0f) * (1.0f / 255.0f);
            ob[0 * 5 + f] = v;
            ob[1 * 5 + f] = v;
            ob[2 * 5 + f] = v;
        }
    }
}

extern "C" void kernel_launch(void* const* d_in, const int* in_sizes, int n_in,
                              void* d_out, int out_size, void* d_ws, size_t ws_size,
                              hipStream_t stream) {
    (void)in_sizes; (void)n_in; (void)out_size; (void)d_ws; (void)ws_size;
    const float* img = (const float*)d_in[0];
    float* out = (float*)d_out;
    glcm_texture_kernel<<<64, BLOCK, SMEM_BYTES, stream>>>(img, out);
}